// GNNModel_884763263854
// MI455X (gfx1250) — compile-verified
//
#include <hip/hip_runtime.h>

// ---------------------------------------------------------------------------
// GNN (2x GCNConv + 3-layer MLP) for MI455X / gfx1250 (wave32, WMMA).
// Memory-bound: edge scatter via L2-resident fp32 atomics; all small GEMMs
// mapped to V_WMMA_F32_16X16X4_F32 (16x16 tile per wave, K padded).
// ---------------------------------------------------------------------------

typedef __attribute__((ext_vector_type(2))) float v2f;
typedef __attribute__((ext_vector_type(8))) float v8f;

#define N_NODES 100000
#define N_EDGES 5000000
#define N_MATCH 1000000
#define N_TILES_NODE (N_NODES / 16)   // 6250
#define N_TILES_MATCH (N_MATCH / 16)  // 62500

__device__ __forceinline__ float leaky(float x) { return x > 0.0f ? x : 0.01f * x; }

// D = A(16x4 f32) * B(4x16 f32) + C(16x16 f32), one tile per wave32.
__device__ __forceinline__ v8f wmma4(v2f a, v2f b, v8f c) {
  return __builtin_amdgcn_wmma_f32_16x16x4_f32(
      /*neg_a=*/false, a, /*neg_b=*/false, b,
      /*c_mod=*/(short)0, c, /*reuse_a=*/false, /*reuse_b=*/false);
}

// --------------------------- utility kernels -------------------------------

__global__ void k_zero(float* p, int n) {
  int i = blockIdx.x * blockDim.x + threadIdx.x;
  if (i < n) p[i] = 0.0f;
}

// deg[dst] += 1 over all edges (L2-resident atomic histogram).
__global__ void k_deg(const int* __restrict__ dst, float* __restrict__ deg) {
  int e = blockIdx.x * blockDim.x + threadIdx.x;
  if (e < N_EDGES) atomicAdd(&deg[dst[e]], 1.0f);
}

// in-place: deg -> rsqrt(deg + 1)  (self-loop)
__global__ void k_dinv(float* __restrict__ deg) {
  int i = blockIdx.x * blockDim.x + threadIdx.x;
  if (i < N_NODES) deg[i] = rsqrtf(deg[i] + 1.0f);
}

// --------------------------- GCN layer 1 -----------------------------------

// hw1[node][0:16] = emb[x[node]] (1x3) @ W1 (3x16), via WMMA K=4 (padded).
// One 16-node tile per wave.
__global__ void k_lin1(const int* __restrict__ x, const float* __restrict__ emb,
                       const float* __restrict__ W1, float* __restrict__ hw1) {
  int wave = (blockIdx.x * blockDim.x + threadIdx.x) >> 5;
  int lane = threadIdx.x & 31;
  if (wave >= N_TILES_NODE) return;
  int m = lane & 15;          // A row == B col == D col for this lane group
  int kb = (lane >> 4) * 2;   // K base: lanes 0-15 -> {0,1}, 16-31 -> {2,3}
  int hi = lane >> 4;

  int node = wave * 16 + m;
  int team = x[node];

  v2f a, b;
  a.x = (kb < 3) ? emb[team * 3 + kb] : 0.0f;
  a.y = (kb + 1 < 3) ? emb[team * 3 + kb + 1] : 0.0f;
  b.x = (kb < 3) ? W1[kb * 16 + m] : 0.0f;
  b.y = (kb + 1 < 3) ? W1[(kb + 1) * 16 + m] : 0.0f;

  v8f c = {};
  c = wmma4(a, b, c);

#pragma unroll
  for (int v = 0; v < 8; ++v) {
    int row = v + 8 * hi;  // D: VGPR v holds rows v (lanes 0-15) and v+8 (16-31)
    hw1[(wave * 16 + row) * 16 + m] = c[v];
  }
}

// Scatter-add 16-wide messages: agg1[dst] += hw1[src] * dinv[src]*dinv[dst]
__global__ void k_edge16(const int* __restrict__ src, const int* __restrict__ dst,
                         const float* __restrict__ dinv, const float* __restrict__ hw1,
                         float* __restrict__ agg1) {
  int e = blockIdx.x * blockDim.x + threadIdx.x;
  if (e >= N_EDGES) return;
  int s = src[e], d = dst[e];
  float norm = dinv[s] * dinv[d];
  const float4* hp = (const float4*)(hw1 + (size_t)s * 16);
  float* ap = agg1 + (size_t)d * 16;
#pragma unroll
  for (int q = 0; q < 4; ++q) {
    float4 h = hp[q];
    atomicAdd(&ap[q * 4 + 0], h.x * norm);
    atomicAdd(&ap[q * 4 + 1], h.y * norm);
    atomicAdd(&ap[q * 4 + 2], h.z * norm);
    atomicAdd(&ap[q * 4 + 3], h.w * norm);
  }
}

// h1 = leaky(agg1 + hw1 * dinv^2 + b1)   (written in-place into agg1)
__global__ void k_post1(const float* __restrict__ dinv, const float* __restrict__ hw1,
                        const float* __restrict__ b1, float* __restrict__ agg1) {
  int i = blockIdx.x * blockDim.x + threadIdx.x;
  if (i >= N_NODES * 16) return;
  int node = i >> 4;
  int j = i & 15;
  float di = dinv[node];
  agg1[i] = leaky(agg1[i] + hw1[i] * di * di + b1[j]);
}

// --------------------------- GCN layer 2 -----------------------------------

// hw2[node][0:3] = h1[node][0:16] @ W2 (16x3, N padded to 16). 4 chained WMMAs.
__global__ void k_lin2(const float* __restrict__ h1, const float* __restrict__ W2,
                       float* __restrict__ hw2) {
  int wave = (blockIdx.x * blockDim.x + threadIdx.x) >> 5;
  int lane = threadIdx.x & 31;
  if (wave >= N_TILES_NODE) return;
  int m = lane & 15;
  int kb = (lane >> 4) * 2;
  int hi = lane >> 4;

  v8f c = {};
#pragma unroll
  for (int kc = 0; kc < 4; ++kc) {
    int k0 = kc * 4 + kb;
    v2f a, b;
    a.x = h1[(wave * 16 + m) * 16 + k0];
    a.y = h1[(wave * 16 + m) * 16 + k0 + 1];
    b.x = (m < 3) ? W2[k0 * 3 + m] : 0.0f;
    b.y = (m < 3) ? W2[(k0 + 1) * 3 + m] : 0.0f;
    c = wmma4(a, b, c);
  }

#pragma unroll
  for (int v = 0; v < 8; ++v) {
    int row = v + 8 * hi;
    if (m < 3) hw2[(wave * 16 + row) * 3 + m] = c[v];
  }
}

// Scatter-add 3-wide messages.
__global__ void k_edge3(const int* __restrict__ src, const int* __restrict__ dst,
                        const float* __restrict__ dinv, const float* __restrict__ hw2,
                        float* __restrict__ agg2) {
  int e = blockIdx.x * blockDim.x + threadIdx.x;
  if (e >= N_EDGES) return;
  int s = src[e], d = dst[e];
  float norm = dinv[s] * dinv[d];
  const float* hp = hw2 + (size_t)s * 3;
  float* ap = agg2 + (size_t)d * 3;
  atomicAdd(&ap[0], hp[0] * norm);
  atomicAdd(&ap[1], hp[1] * norm);
  atomicAdd(&ap[2], hp[2] * norm);
}

// h2 = leaky(agg2 + hw2 * dinv^2 + b2)   (in-place into agg2)
__global__ void k_post2(const float* __restrict__ dinv, const float* __restrict__ hw2,
                        const float* __restrict__ b2, float* __restrict__ agg2) {
  int i = blockIdx.x * blockDim.x + threadIdx.x;
  if (i >= N_NODES * 3) return;
  int node = i / 3;
  int j = i % 3;
  float di = dinv[node];
  agg2[i] = leaky(agg2[i] + hw2[i] * di * di + b2[j]);
}

// --------------------------- fused MLP head --------------------------------

// Per wave: 16 matches. z = [h2[home], h2[away]] (16x6) -> 16 -> 6 -> 3.
// All three layers as f32 WMMA (K padded); inter-layer 16x16 activations
// staged through a padded LDS tile to convert D-layout -> A-layout.
__global__ void __launch_bounds__(128)
k_mlp(const float* __restrict__ h2, const int* __restrict__ home,
      const int* __restrict__ away,
      const float* __restrict__ lw1, const float* __restrict__ lb1,
      const float* __restrict__ lw2, const float* __restrict__ lb2,
      const float* __restrict__ lw3, const float* __restrict__ lb3,
      float* __restrict__ out) {
  __shared__ float lds[4][16][17];  // 4 waves/block, +1 col pad vs bank conflicts

  int w = threadIdx.x >> 5;
  int lane = threadIdx.x & 31;
  int tile = blockIdx.x * 4 + w;  // grid sized so every tile is valid
  int m = lane & 15;
  int kb = (lane >> 4) * 2;
  int hi = lane >> 4;

  int row = tile * 16 + m;
  int hm = home[row];
  int aw = away[row];

  // ----- layer 1: (16x6 pad 8) @ lw1 (6x16 pad 8x16), 2 chained WMMAs -----
  v8f c1 = {};
#pragma unroll
  for (int kc = 0; kc < 2; ++kc) {
    int k0 = kc * 4 + kb;
    v2f a, b;
    float a0 = (k0 < 3) ? h2[hm * 3 + k0] : ((k0 < 6) ? h2[aw * 3 + (k0 - 3)] : 0.0f);
    int k1 = k0 + 1;
    float a1 = (k1 < 3) ? h2[hm * 3 + k1] : ((k1 < 6) ? h2[aw * 3 + (k1 - 3)] : 0.0f);
    a.x = a0;
    a.y = a1;
    b.x = (k0 < 6) ? lw1[k0 * 16 + m] : 0.0f;
    b.y = (k1 < 6) ? lw1[k1 * 16 + m] : 0.0f;
    c1 = wmma4(a, b, c1);
  }
  float bias1 = lb1[m];
#pragma unroll
  for (int v = 0; v < 8; ++v) {
    int r = v + 8 * hi;
    lds[w][r][m] = leaky(c1[v] + bias1);
  }
  __syncthreads();

  // ----- layer 2: (16x16) @ lw2 (16x6 pad to 16 cols), 4 chained WMMAs -----
  v8f c2 = {};
#pragma unroll
  for (int kc = 0; kc < 4; ++kc) {
    int k0 = kc * 4 + kb;
    v2f a, b;
    a.x = lds[w][m][k0];
    a.y = lds[w][m][k0 + 1];
    b.x = (m < 6) ? lw2[k0 * 6 + m] : 0.0f;
    b.y = (m < 6) ? lw2[(k0 + 1) * 6 + m] : 0.0f;
    c2 = wmma4(a, b, c2);
  }
  __syncthreads();  // done reading layer-1 tile before overwrite
  float bias2 = (m < 6) ? lb2[m] : 0.0f;
#pragma unroll
  for (int v = 0; v < 8; ++v) {
    int r = v + 8 * hi;
    lds[w][r][m] = leaky(c2[v] + bias2);  // cols >= 6 are exactly 0
  }
  __syncthreads();

  // ----- layer 3: (16x6 pad 8) @ lw3 (6x3 pad), 2 chained WMMAs -----
  v8f c3 = {};
#pragma unroll
  for (int kc = 0; kc < 2; ++kc) {
    int k0 = kc * 4 + kb;
    int k1 = k0 + 1;
    v2f a, b;
    a.x = (k0 < 6) ? lds[w][m][k0] : 0.0f;
    a.y = (k1 < 6) ? lds[w][m][k1] : 0.0f;
    b.x = (k0 < 6 && m < 3) ? lw3[k0 * 3 + m] : 0.0f;
    b.y = (k1 < 6 && m < 3) ? lw3[k1 * 3 + m] : 0.0f;
    c3 = wmma4(a, b, c3);
  }
#pragma unroll
  for (int v = 0; v < 8; ++v) {
    int r = v + 8 * hi;
    if (m < 3) out[(size_t)(tile * 16 + r) * 3 + m] = leaky(c3[v] + lb3[m]);
  }
}

// --------------------------- launcher --------------------------------------

extern "C" void kernel_launch(void* const* d_in, const int* in_sizes, int n_in,
                              void* d_out, int out_size, void* d_ws, size_t ws_size,
                              hipStream_t stream) {
  const int* x = (const int*)d_in[0];
  const int* eidx = (const int*)d_in[1];
  const int* src = eidx;             // edge_index[0]
  const int* dst = eidx + N_EDGES;   // edge_index[1]
  const int* home = (const int*)d_in[2];
  const int* away = (const int*)d_in[3];
  const float* emb = (const float*)d_in[4];
  const float* W1 = (const float*)d_in[5];
  const float* b1 = (const float*)d_in[6];
  const float* W2 = (const float*)d_in[7];
  const float* b2 = (const float*)d_in[8];
  const float* lw1 = (const float*)d_in[9];
  const float* lb1 = (const float*)d_in[10];
  const float* lw2 = (const float*)d_in[11];
  const float* lb2 = (const float*)d_in[12];
  const float* lw3 = (const float*)d_in[13];
  const float* lb3 = (const float*)d_in[14];
  float* out = (float*)d_out;

  // Workspace layout (floats): deg/dinv | agg1(16N) | agg2(3N) | hw1(16N) | hw2(3N)
  float* ws = (float*)d_ws;
  float* deg = ws;                     // N      (becomes dinv in-place)
  float* agg1 = ws + N_NODES;          // 16N    (becomes h1 in-place)
  float* agg2 = ws + 17 * N_NODES;     // 3N     (becomes h2 in-place)
  float* hw1 = ws + 20 * N_NODES;      // 16N
  float* hw2 = ws + 36 * N_NODES;      // 3N

  const int T = 256;
  int zn = 20 * N_NODES;  // deg + agg1 + agg2 must start at zero
  k_zero<<<(zn + T - 1) / T, T, 0, stream>>>(ws, zn);

  k_deg<<<(N_EDGES + T - 1) / T, T, 0, stream>>>(dst, deg);
  k_dinv<<<(N_NODES + T - 1) / T, T, 0, stream>>>(deg);

  int linBlocks = (N_TILES_NODE * 32 + T - 1) / T;  // 1 tile per wave32
  k_lin1<<<linBlocks, T, 0, stream>>>(x, emb, W1, hw1);
  k_edge16<<<(N_EDGES + T - 1) / T, T, 0, stream>>>(src, dst, deg, hw1, agg1);
  k_post1<<<(N_NODES * 16 + T - 1) / T, T, 0, stream>>>(deg, hw1, b1, agg1);

  k_lin2<<<linBlocks, T, 0, stream>>>(agg1, W2, hw2);
  k_edge3<<<(N_EDGES + T - 1) / T, T, 0, stream>>>(src, dst, deg, hw2, agg2);
  k_post2<<<(N_NODES * 3 + T - 1) / T, T, 0, stream>>>(deg, hw2, b2, agg2);

  // 62500 tiles, 4 waves (tiles) per 128-thread block -> 15625 blocks, all valid.
  k_mlp<<<N_TILES_MATCH / 4, 128, 0, stream>>>(agg2, home, away, lw1, lb1, lw2, lb2,
                                               lw3, lb3, out);
}